// PointCloudModel_8220567405261
// MI455X (gfx1250) — compile-verified
//
#include <hip/hip_runtime.h>
#include <hip/hip_bf16.h>

// FINAL (converged, round 3): streaming point-cloud projection for MI455X
// (gfx1250, wave32).
//
// Rooflines: ~100 B/point mandatory HBM traffic -> ~8.6 us floor @ 23.3 TB/s;
// VALU at 178 instr/point -> ~4.4 us across ~1024 SIMD32s, fully hidden.
// Bandwidth-bound, so the optimized shape is data movement, not matrix ops:
//  - one point per wave32 lane, every byte of every touched line consumed
//  - uniform W/K via s_load_b256 scalar clauses
//  - global_prefetch_b8 run-ahead into GL2 for both input streams
//  - nontemporal (TH=NT) stores for the write-once 104 MB output stream
//  - symmetric-matrix math (6 of 9 elements) + v_rcp_f32 instead of IEEE div
// WMMA is deliberately absent: the only uniform linear maps (R (x) R, K (x) K on
// the 9-vector of covariance entries) are <0.5% of the time budget and would
// require cross-lane VGPR layout shuffles costing more VALU than they remove.

#define PC_BLOCK 256
#define PC_PF_DIST 2048  // points of run-ahead (~72 KB of cov stream per step)

__global__ __launch_bounds__(PC_BLOCK) void pc_project_kernel(
    const float* __restrict__ coords,   // (N,3)
    const float* __restrict__ cov,      // (N,3,3)
    const float* __restrict__ W,        // (4,4)
    const float* __restrict__ Kmat,     // (3,3)
    float* __restrict__ out_img,        // (N,3)
    float* __restrict__ out_cov,        // (N,3,3)
    float* __restrict__ out_mask,       // (N,)
    int n)
{
    int i = blockIdx.x * PC_BLOCK + threadIdx.x;
    if (i >= n) return;
    size_t ib = (size_t)i;

    // ---- uniform camera extrinsics / intrinsics (lowered to s_load clauses) ----
    float R[3][3], t[3], K[3][3];
#pragma unroll
    for (int r = 0; r < 3; ++r) {
#pragma unroll
        for (int c = 0; c < 3; ++c) R[r][c] = W[r * 4 + c];
        t[r] = W[r * 4 + 3];
    }
#pragma unroll
    for (int r = 0; r < 3; ++r)
#pragma unroll
        for (int c = 0; c < 3; ++c) K[r][c] = Kmat[r * 3 + c];

    // ---- prefetch run-ahead into GL2 (global_prefetch_b8; lanes cover every
    //      128B line of both streams at +PC_PF_DIST points) ----
    if (i + PC_PF_DIST < n) {
        __builtin_prefetch(cov + (ib + PC_PF_DIST) * 9, 0, 0);
        __builtin_prefetch(coords + (ib + PC_PF_DIST) * 3, 0, 0);
    }

    // ---- per-point loads ----
    const float p0 = coords[ib * 3 + 0];
    const float p1 = coords[ib * 3 + 1];
    const float p2 = coords[ib * 3 + 2];

    float C[3][3];
#pragma unroll
    for (int r = 0; r < 3; ++r)
#pragma unroll
        for (int c = 0; c < 3; ++c) C[r][c] = cov[ib * 9 + r * 3 + c];

    // ---- cam = R p + t ----
    float cam[3];
#pragma unroll
    for (int r = 0; r < 3; ++r)
        cam[r] = R[r][0] * p0 + R[r][1] * p1 + R[r][2] * p2 + t[r];

    const bool m = (cam[0] > 0.0f) && (cam[1] > 0.0f) && (cam[2] > 0.0f);

    // ---- cam_cov = R C R^T (symmetric: 6 unique elements) ----
    float T[3][3];
#pragma unroll
    for (int r = 0; r < 3; ++r)
#pragma unroll
        for (int c = 0; c < 3; ++c)
            T[r][c] = R[r][0] * C[0][c] + R[r][1] * C[1][c] + R[r][2] * C[2][c];

    const float cc00 = T[0][0] * R[0][0] + T[0][1] * R[0][1] + T[0][2] * R[0][2];
    const float cc01 = T[0][0] * R[1][0] + T[0][1] * R[1][1] + T[0][2] * R[1][2];
    const float cc02 = T[0][0] * R[2][0] + T[0][1] * R[2][1] + T[0][2] * R[2][2];
    const float cc11 = T[1][0] * R[1][0] + T[1][1] * R[1][1] + T[1][2] * R[1][2];
    const float cc12 = T[1][0] * R[2][0] + T[1][1] * R[2][1] + T[1][2] * R[2][2];
    const float cc22 = T[2][0] * R[2][0] + T[2][1] * R[2][1] + T[2][2] * R[2][2];

    // ---- perspective Jacobian (sparse form) ----
    const float x = cam[0];
    const float y = cam[1];
    const float z = m ? cam[2] : 1.0f;
    const float iz = __builtin_amdgcn_rcpf(z);   // v_rcp_f32, ~1 ulp; rcp(1)==1 exactly
    const float a = -x * iz * iz;  // J[0][2]
    const float b = -y * iz * iz;  // J[1][2]

    // U = J * cam_cov  (row 2 of J is e3 -> U row 2 == cam_cov row 2)
    const float U00 = iz * cc00 + a * cc02;
    const float U01 = iz * cc01 + a * cc12;
    const float U02 = iz * cc02 + a * cc22;
    const float U11 = iz * cc11 + b * cc12;
    const float U12 = iz * cc12 + b * cc22;

    // ray_cov = U * J^T (symmetric: 6 unique elements)
    const float rc00 = U00 * iz + U02 * a;
    const float rc01 = U01 * iz + U02 * b;
    const float rc02 = U02;
    const float rc11 = U11 * iz + U12 * b;
    const float rc12 = U12;
    const float rc22 = cc22;

    const float RCm[3][3] = {{rc00, rc01, rc02},
                             {rc01, rc11, rc12},
                             {rc02, rc12, rc22}};

    // ---- img = K * ray ; img_cov = K * ray_cov * K^T (symmetric) ----
    const float rx = x * iz;
    const float ry = y * iz;

    float img[3];
#pragma unroll
    for (int r = 0; r < 3; ++r)
        img[r] = K[r][0] * rx + K[r][1] * ry + K[r][2];

    float V[3][3];
#pragma unroll
    for (int r = 0; r < 3; ++r)
#pragma unroll
        for (int c = 0; c < 3; ++c)
            V[r][c] = K[r][0] * RCm[0][c] + K[r][1] * RCm[1][c] + K[r][2] * RCm[2][c];

    const float ic00 = V[0][0] * K[0][0] + V[0][1] * K[0][1] + V[0][2] * K[0][2];
    const float ic01 = V[0][0] * K[1][0] + V[0][1] * K[1][1] + V[0][2] * K[1][2];
    const float ic02 = V[0][0] * K[2][0] + V[0][1] * K[2][1] + V[0][2] * K[2][2];
    const float ic11 = V[1][0] * K[1][0] + V[1][1] * K[1][1] + V[1][2] * K[1][2];
    const float ic12 = V[1][0] * K[2][0] + V[1][1] * K[2][1] + V[1][2] * K[2][2];
    const float ic22 = V[2][0] * K[2][0] + V[2][1] * K[2][1] + V[2][2] * K[2][2];

    const float IC[3][3] = {{ic00, ic01, ic02},
                            {ic01, ic11, ic12},
                            {ic02, ic12, ic22}};

    // ---- masked, nontemporal stores (write-once stream) ----
#pragma unroll
    for (int r = 0; r < 3; ++r)
        __builtin_nontemporal_store(m ? img[r] : 0.0f, &out_img[ib * 3 + r]);
#pragma unroll
    for (int r = 0; r < 3; ++r)
#pragma unroll
        for (int c = 0; c < 3; ++c)
            __builtin_nontemporal_store(m ? IC[r][c] : 0.0f, &out_cov[ib * 9 + r * 3 + c]);
    __builtin_nontemporal_store(m ? 1.0f : 0.0f, &out_mask[ib]);
}

extern "C" void kernel_launch(void* const* d_in, const int* in_sizes, int n_in,
                              void* d_out, int out_size, void* d_ws, size_t ws_size,
                              hipStream_t stream) {
    const float* coords = (const float*)d_in[0];  // (N,3)
    const float* cov    = (const float*)d_in[1];  // (N,3,3)
    const float* W      = (const float*)d_in[2];  // (4,4)
    const float* K      = (const float*)d_in[3];  // (3,3)

    const int n = in_sizes[0] / 3;

    float* out      = (float*)d_out;
    float* out_img  = out;                      // N*3
    float* out_cov  = out + (size_t)n * 3;      // N*9
    float* out_mask = out + (size_t)n * 12;     // N

    const int grid = (n + PC_BLOCK - 1) / PC_BLOCK;
    pc_project_kernel<<<grid, PC_BLOCK, 0, stream>>>(coords, cov, W, K,
                                                     out_img, out_cov, out_mask, n);
}